// PositiveGraphEncoder_89352499626208
// MI455X (gfx1250) — compile-verified
//
#include <hip/hip_runtime.h>
#include <hip/hip_bf16.h>
#include <math.h>

#define NN  100000
#define MM  3
#define DIN 128
#define DH  64
#define NE  1600000

typedef __attribute__((ext_vector_type(2))) float v2f;
typedef __attribute__((ext_vector_type(4))) float v4f;
typedef __attribute__((ext_vector_type(8))) float v8f;

// workspace layout (in floats)
#define AGG_OFF   0ull
#define DEG_OFF   19200000ull   // MM*NN*DH
#define LOG_OFF   19500000ull   // + MM*NN
#define BETA_OFF  19500004ull
#define PROJ_OFF  19500008ull

// Native CDNA5 L2 atomic f32 add, no-return form (STOREcnt, fire-and-forget).
// Device scope: aggregation rows are shared across WGPs.
__device__ __forceinline__ void atomicAddF(float* p, float v) {
  asm volatile("global_atomic_add_f32 %0, %1, off scope:SCOPE_DEV"
               :: "v"(p), "v"(v) : "memory");
}

// ---------------- Kernel 1: proj[m,n,h] = feats[m,n,:] @ W[m,:,h]  (WMMA f32 16x16x4)
__global__ __launch_bounds__(128)
void k_proj_wmma(const float* __restrict__ feats, const float* __restrict__ W,
                 float* __restrict__ proj) {
  const int lane = threadIdx.x & 31;
  const int wv   = threadIdx.x >> 5;     // 4 waves -> 4 col tiles of 16
  const int col0 = wv * 16;
  const int m    = blockIdx.y;
  const int n0   = blockIdx.x * 16;      // NN = 16*6250 exactly
  const int r    = lane & 15;            // row within tile (A) / col within tile (B)
  const int hi   = lane >> 4;            // 0: K pair {0,1}, 1: K pair {2,3}
  const int kh   = hi * 2;

  const float* A = feats + ((size_t)m * NN + n0) * DIN;
  const float* B = W + (size_t)m * DIN * DH;

  v8f acc = {0.f,0.f,0.f,0.f,0.f,0.f,0.f,0.f};
  #pragma unroll
  for (int k = 0; k < DIN; k += 4) {
    v2f a = *(const v2f*)(A + (size_t)r * DIN + k + kh);   // A[r][k+kh], A[r][k+kh+1]
    v2f b;
    b.x = B[(k + kh)     * DH + col0 + r];                 // B[k+kh][col]
    b.y = B[(k + kh + 1) * DH + col0 + r];
    acc = __builtin_amdgcn_wmma_f32_16x16x4_f32(false, a, false, b, (short)0, acc,
                                                false, false);
  }
  float* O = proj + ((size_t)m * NN + n0) * DH;
  #pragma unroll
  for (int v = 0; v < 8; ++v)                              // row = v + hi*8, col = col0+r
    O[(size_t)(v + hi * 8) * DH + col0 + r] = acc[v];
}

// ---------------- Kernel 2: edge scatter-add (segment_sum) + degree
__global__ __launch_bounds__(256)
void k_scatter(const float* __restrict__ proj, const int* __restrict__ src,
               const int* __restrict__ dst, float* __restrict__ agg,
               float* __restrict__ deg) {
  size_t t = (size_t)blockIdx.x * 256 + threadIdx.x;
  if (t >= (size_t)MM * NE * DH) return;
  int    j  = (int)(t & (DH - 1));
  size_t e  = t >> 6;                       // [0, MM*NE) == m*NE + ei
  int    m  = (int)(e / NE);
  int s = src[e];
  int d = dst[e];
  float val = proj[((size_t)m * NN + s) * DH + j];
  atomicAddF(&agg[((size_t)m * NN + d) * DH + j], val);
  if (j == 0) atomicAddF(&deg[(size_t)m * NN + d], 1.0f);
}

// ---------------- Kernel 3: h = prelu(agg/max(deg,1) + b), in place, float4
__global__ __launch_bounds__(256)
void k_prelu(float* __restrict__ agg, const float* __restrict__ deg,
             const float* __restrict__ b, const float* __restrict__ pa) {
  size_t i4 = (size_t)blockIdx.x * 256 + threadIdx.x;     // over MM*NN*(DH/4)
  if (i4 >= (size_t)MM * NN * (DH / 4)) return;
  size_t base = i4 * 4;
  size_t nl   = i4 >> 4;                    // m*NN + n
  int m  = (int)(nl / NN);
  int j4 = (int)(i4 & 15);                  // float4 group within the row
  float d = fmaxf(deg[nl], 1.0f);
  float a = pa[m];
  v4f hv = *(v4f*)(agg + base);
  v4f bv = *(const v4f*)(b + m * DH + j4 * 4);
  #pragma unroll
  for (int c = 0; c < 4; ++c) {
    float h = hv[c] / d + bv[c];
    hv[c] = h > 0.f ? h : a * h;
  }
  *(v4f*)(agg + base) = hv;
}

// ---------------- Kernel 4: s = tanh(h @ fc_W + fc_b); logits[m] += s . attn  (WMMA)
__global__ __launch_bounds__(128)
void k_fc_wmma(const float* __restrict__ h, const float* __restrict__ fcW,
               const float* __restrict__ fcb, const float* __restrict__ attn,
               float* __restrict__ logits) {
  __shared__ float red[128];
  const int lane = threadIdx.x & 31;
  const int wv   = threadIdx.x >> 5;
  const int col0 = wv * 16;
  const int m    = blockIdx.y;
  const int n0   = blockIdx.x * 16;
  const int r    = lane & 15;
  const int hi   = lane >> 4;
  const int kh   = hi * 2;

  const float* A = h + ((size_t)m * NN + n0) * DH;
  float bias = fcb[col0 + r];
  v8f acc = {bias, bias, bias, bias, bias, bias, bias, bias};
  #pragma unroll
  for (int k = 0; k < DH; k += 4) {
    v2f a = *(const v2f*)(A + (size_t)r * DH + k + kh);
    v2f bb;
    bb.x = fcW[(k + kh)     * DH + col0 + r];
    bb.y = fcW[(k + kh + 1) * DH + col0 + r];
    acc = __builtin_amdgcn_wmma_f32_16x16x4_f32(false, a, false, bb, (short)0, acc,
                                                false, false);
  }
  float p = 0.f;
  #pragma unroll
  for (int v = 0; v < 8; ++v) p += tanhf(acc[v]);
  p *= attn[col0 + r];                      // col index == h index of s

  red[threadIdx.x] = p;
  __syncthreads();
  #pragma unroll
  for (int s2 = 64; s2 > 0; s2 >>= 1) {
    if (threadIdx.x < s2) red[threadIdx.x] += red[threadIdx.x + s2];
    __syncthreads();
  }
  if (threadIdx.x == 0) atomicAddF(&logits[m], red[0]);
}

// ---------------- Kernel 5: beta = softmax(logits / NN)
__global__ void k_softmax(const float* __restrict__ logits, float* __restrict__ beta) {
  if (threadIdx.x == 0 && blockIdx.x == 0) {
    float l[MM]; float mx = -1e30f;
    for (int m = 0; m < MM; ++m) { l[m] = logits[m] / (float)NN; mx = fmaxf(mx, l[m]); }
    float s = 0.f;
    for (int m = 0; m < MM; ++m) { l[m] = expf(l[m] - mx); s += l[m]; }
    for (int m = 0; m < MM; ++m) beta[m] = l[m] / s;
  }
}

// ---------------- Kernel 6: z[n,h] = sum_m beta[m] * h[m,n,h], float4
__global__ __launch_bounds__(256)
void k_out(const float* __restrict__ h, const float* __restrict__ beta,
           float* __restrict__ out) {
  size_t i4 = (size_t)blockIdx.x * 256 + threadIdx.x;     // over NN*(DH/4)
  if (i4 >= (size_t)NN * (DH / 4)) return;
  size_t base = i4 * 4;
  const size_t P = (size_t)NN * DH;
  float b0 = beta[0], b1 = beta[1], b2 = beta[2];
  v4f x0 = *(const v4f*)(h + base);
  v4f x1 = *(const v4f*)(h + P + base);
  v4f x2 = *(const v4f*)(h + 2 * P + base);
  v4f z  = x0 * b0 + x1 * b1 + x2 * b2;
  *(v4f*)(out + base) = z;
}

extern "C" void kernel_launch(void* const* d_in, const int* in_sizes, int n_in,
                              void* d_out, int out_size, void* d_ws, size_t ws_size,
                              hipStream_t stream) {
  const float* feats = (const float*)d_in[0];
  const int*   src   = (const int*)  d_in[1];
  const int*   dst   = (const int*)  d_in[2];
  const float* W     = (const float*)d_in[3];
  const float* b     = (const float*)d_in[4];
  const float* pa    = (const float*)d_in[5];
  const float* fcW   = (const float*)d_in[6];
  const float* fcb   = (const float*)d_in[7];
  const float* attn  = (const float*)d_in[8];

  float* ws     = (float*)d_ws;
  float* agg    = ws + AGG_OFF;
  float* deg    = ws + DEG_OFF;
  float* logits = ws + LOG_OFF;
  float* beta   = ws + BETA_OFF;
  float* proj   = ws + PROJ_OFF;
  float* out    = (float*)d_out;

  // zero agg + deg + logits (+beta slot) every call: deterministic accumulators
  hipMemsetAsync(d_ws, 0, (size_t)(LOG_OFF + 8) * sizeof(float), stream);

  dim3 gTiles(NN / 16, MM);                 // 6250 x 3 tiles of 16 rows
  k_proj_wmma<<<gTiles, 128, 0, stream>>>(feats, W, proj);

  size_t total = (size_t)MM * NE * DH;      // 307.2M lanes: 64 lanes per edge
  unsigned blocks = (unsigned)((total + 255) / 256);
  k_scatter<<<blocks, 256, 0, stream>>>(proj, src, dst, agg, deg);

  size_t nv4 = (size_t)MM * NN * (DH / 4);
  k_prelu<<<(unsigned)((nv4 + 255) / 256), 256, 0, stream>>>(agg, deg, b, pa);

  k_fc_wmma<<<gTiles, 128, 0, stream>>>(agg, fcW, fcb, attn, logits);

  k_softmax<<<1, 32, 0, stream>>>(logits, beta);

  size_t ov4 = (size_t)NN * (DH / 4);
  k_out<<<(unsigned)((ov4 + 255) / 256), 256, 0, stream>>>(agg, beta, out);
}